// KMeans_21019569947069
// MI455X (gfx1250) — compile-verified
//
#include <hip/hip_runtime.h>

#define KB 8
#define KN 65536
#define KC 512
#define KD 128
#define KITERS 10
#define UPD_BLOCKS 32

typedef float v2f __attribute__((ext_vector_type(2)));
typedef float v8f __attribute__((ext_vector_type(8)));
typedef unsigned int u32x4 __attribute__((ext_vector_type(4)));
typedef int i32x8 __attribute__((ext_vector_type(8)));
typedef int i32x4 __attribute__((ext_vector_type(4)));

// ---------------------------------------------------------------------------
// TDM: DMA the whole [KC][KD] f32 center block (256KB) into LDS with one
// tensor_load_to_lds (2D descriptor). Tracked by TENSORcnt.
// D# packing per CDNA5 ISA.
// This toolchain declares the 6-arg builtin:
//   (u32x4 g0, i32x8 g1, i32x4 g2, i32x4 g3, i32x8 g4, i32 cpol)
// ---------------------------------------------------------------------------
#if __has_builtin(__builtin_amdgcn_tensor_load_to_lds)
#define HAVE_TDM 1
static __device__ __forceinline__ void tdm_load_centers(const float* gsrc, float* lds_dst)
{
    unsigned long long ga = (unsigned long long)(uintptr_t)gsrc;
    unsigned int lo = (unsigned int)(uintptr_t)lds_dst;   // LDS byte offset

    u32x4 g0;
    g0[0] = 1u;                                    // count=1, user-mode, no gather
    g0[1] = lo;                                    // lds_addr [63:32]
    g0[2] = (unsigned int)(ga & 0xFFFFFFFFu);      // global_addr [95:64]
    g0[3] = (unsigned int)((ga >> 32) & 0x01FFFFFFu) // global_addr [120:96]
          | (2u << 30);                            // type=2 ("image") [127:126]

    i32x8 g1;
    g1[0] = (int)(2u << 16);        // workgroup_mask=0, data_size=2 (4 bytes)
    g1[1] = (int)(128u << 16);      // atomic_barrier_addr=0 | tensor_dim0.lo16 (=128)
    g1[2] = (int)(512u << 16);      // tensor_dim0.hi16 (=0) | tensor_dim1.lo16 (=512)
    g1[3] = (int)(128u << 16);      // tensor_dim1.hi16 (=0) | tile_dim0=128
    g1[4] = (int)512;               // tile_dim1=512 | tile_dim2=0
    g1[5] = (int)128;               // tensor_dim0_stride.lo32 = 128 elements
    g1[6] = 0;                      // stride0.hi16 | tensor_dim1_stride.lo16
    g1[7] = 0;                      // tensor_dim1_stride.hi32

    i32x4 z4 = {0, 0, 0, 0};
    i32x8 z8 = {0, 0, 0, 0, 0, 0, 0, 0};
    __builtin_amdgcn_tensor_load_to_lds(g0, g1, z4, z4, z8, 0);
}
#else
#define HAVE_TDM 0
#endif

// ---------------------------------------------------------------------------
// Assignment: per batch, G = Centers(16xK tiles) x Data(Kx16 tiles) via
// V_WMMA_F32_16X16X4_F32; argmin_c( ||c||^2 - 2*G[c][n] ) per data row.
// All 512 centers (256KB) + norms staged in LDS (CDNA5: 320KB per WGP).
// ---------------------------------------------------------------------------
__global__ __launch_bounds__(256, 1)
void kmeans_assign(const float* __restrict__ data,
                   const float* __restrict__ centers,
                   const float* __restrict__ cnorm,
                   int* __restrict__ assign)
{
    extern __shared__ float smem[];          // [KC*KD] centers + [KC] norms
    float* sC = smem;
    float* sN = smem + KC * KD;

    const int b    = blockIdx.y;
    const int tid  = threadIdx.x;
    const int lane = tid & 31;
    const int wave = tid >> 5;
    const int hi   = lane >> 4;              // 0: K+{0,1}, 1: K+{2,3}
    const int lr   = lane & 15;

#if HAVE_TDM
    // One TDM DMA per block (TDM ignores EXEC; uniform branch -> 1 issue).
    if (tid < 32) {
        tdm_load_centers(centers + (size_t)b * KC * KD, sC);
    }
#else
    {
        const float4* gc4 = (const float4*)(centers + (size_t)b * KC * KD);
        float4* sc4 = (float4*)sC;
        for (int i = tid; i < (KC * KD) / 4; i += 256) sc4[i] = gc4[i];
    }
#endif
    for (int i = tid; i < KC; i += 256) sN[i] = cnorm[b * KC + i];

    // This wave's 16 data rows, resident in registers in B-matrix layout:
    // lane n (n<16) holds X[row n][4kk+0..1], lane n+16 holds X[row n][4kk+2..3].
    const int row0 = blockIdx.x * 128 + wave * 16;
    const float* xr = data + ((size_t)b * KN + row0 + lr) * KD + 2 * hi;
    __builtin_prefetch(xr, 0, 0);            // global_prefetch_b8
    v2f Bv[32];
#pragma unroll
    for (int kk = 0; kk < 32; ++kk)
        Bv[kk] = *(const v2f*)(xr + 4 * kk);

#if HAVE_TDM
    if (tid < 32) {
        __builtin_amdgcn_s_wait_tensorcnt(0);   // LDS tile complete before signal
    }
#endif
    __syncthreads();

    float best  = 3.4e38f;
    int   bestc = 0;

#pragma unroll 1
    for (int t = 0; t < KC / 16; ++t) {
        // A-matrix (centers tile) from LDS: lane l holds center row (l&15),
        // K pairs split across lane halves — symmetric with B layout.
        const float* cr = sC + (t * 16 + lr) * KD + 2 * hi;
        v8f acc0 = {0.f, 0.f, 0.f, 0.f, 0.f, 0.f, 0.f, 0.f};
        v8f acc1 = {0.f, 0.f, 0.f, 0.f, 0.f, 0.f, 0.f, 0.f};
        // software-pipelined A loads (one k-step ahead of the WMMAs)
        v2f a0 = *(const v2f*)(cr + 0);
        v2f a1 = *(const v2f*)(cr + 4);
#pragma unroll
        for (int kk = 0; kk < 32; kk += 2) {
            v2f n0, n1;
            if (kk + 2 < 32) {
                n0 = *(const v2f*)(cr + 4 * (kk + 2));
                n1 = *(const v2f*)(cr + 4 * (kk + 2) + 4);
            }
            acc0 = __builtin_amdgcn_wmma_f32_16x16x4_f32(
                false, a0, false, Bv[kk],     (short)0, acc0, false, false);
            acc1 = __builtin_amdgcn_wmma_f32_16x16x4_f32(
                false, a1, false, Bv[kk + 1], (short)0, acc1, false, false);
            a0 = n0; a1 = n1;
        }
        v8f acc = acc0 + acc1;
        // D layout: lane (half h) vgpr m -> M = m + 8*h (center), N = lr (row).
#pragma unroll
        for (int m = 0; m < 8; ++m) {
            int   c = t * 16 + m + 8 * hi;
            float v = sN[c] - 2.0f * acc[m];
            if (v < best) { best = v; bestc = c; }   // ascending c => first-min
        }
    }

    // Merge the two lane halves (same data row, disjoint center subsets).
    float ob = __shfl_xor(best, 16, 32);
    int   oc = __shfl_xor(bestc, 16, 32);
    if (ob < best || (ob == best && oc < bestc)) { best = ob; bestc = oc; }
    if (hi == 0)
        assign[(size_t)b * KN + row0 + lr] = bestc;
}

// ---------------------------------------------------------------------------
// Center norms ||c||^2 per (b,c).
// ---------------------------------------------------------------------------
__global__ void kmeans_cnorm(const float* __restrict__ centers,
                             float* __restrict__ cnorm)
{
    int i = blockIdx.x * blockDim.x + threadIdx.x;
    if (i >= KB * KC) return;
    const float* c = centers + (size_t)i * KD;
    float s = 0.f;
#pragma unroll 4
    for (int d = 0; d < KD; ++d) { float v = c[d]; s += v * v; }
    cnorm[i] = s;
}

__global__ void kmeans_zero(float* __restrict__ sums, float* __restrict__ counts)
{
    size_t stride = (size_t)gridDim.x * blockDim.x;
    size_t i0 = (size_t)blockIdx.x * blockDim.x + threadIdx.x;
    for (size_t i = i0; i < (size_t)KB * KC * KD; i += stride) sums[i] = 0.f;
    for (size_t i = i0; i < (size_t)KB * KC; i += stride) counts[i] = 0.f;
}

// ---------------------------------------------------------------------------
// Scatter-mean accumulation: LDS-privatized [C][D] sums (+counts) per block,
// ds_add_f32 bank-conflict-free (lanes stride d by 32), atomic flush.
// ---------------------------------------------------------------------------
__global__ __launch_bounds__(256, 1)
void kmeans_accum(const float* __restrict__ data,
                  const int* __restrict__ assign,
                  float* __restrict__ sums,
                  float* __restrict__ counts)
{
    extern __shared__ float smem[];          // [KC*KD] sums + [KC] counts
    float* lsum = smem;
    float* lcnt = smem + KC * KD;

    const int b    = blockIdx.y;
    const int tid  = threadIdx.x;
    const int lane = tid & 31;
    const int wave = tid >> 5;

    for (int i = tid; i < KC * KD; i += 256) lsum[i] = 0.f;
    for (int i = tid; i < KC; i += 256) lcnt[i] = 0.f;
    __syncthreads();

    const int PPB = KN / UPD_BLOCKS;         // points per block (2048)
    const int PPW = PPB / 8;                 // points per wave  (256)
    int p0 = blockIdx.x * PPB + wave * PPW;
    for (int i = 0; i < PPW; ++i) {
        int p = p0 + i;
        int a = assign[(size_t)b * KN + p];  // uniform broadcast load
        const float* xp = data + ((size_t)b * KN + p) * KD;
#pragma unroll
        for (int j = 0; j < 4; ++j)
            atomicAdd(&lsum[a * KD + lane + 32 * j], xp[lane + 32 * j]);
        if (lane == 0) atomicAdd(&lcnt[a], 1.0f);
    }
    __syncthreads();

    for (int i = tid; i < KC * KD; i += 256) {
        float v = lsum[i];
        if (v != 0.f) atomicAdd(&sums[(size_t)b * KC * KD + i], v);
    }
    for (int i = tid; i < KC; i += 256) {
        float v = lcnt[i];
        if (v != 0.f) atomicAdd(&counts[b * KC + i], v);
    }
}

// new_center = count>0 ? sum/count : old_center   (in-place safe: same thread RW)
__global__ void kmeans_update(const float* __restrict__ sums,
                              const float* __restrict__ counts,
                              float* __restrict__ centers)
{
    int bc = blockIdx.x;                     // over B*C, 128 threads = D
    float cnt = counts[bc];
    size_t o = (size_t)bc * KD + threadIdx.x;
    float oldv = centers[o];
    float s = sums[o];
    centers[o] = (cnt > 0.f) ? (s / cnt) : oldv;
}

__global__ void kmeans_copyc(const float* __restrict__ src, float* __restrict__ dst)
{
    size_t stride = (size_t)gridDim.x * blockDim.x;
    for (size_t i = (size_t)blockIdx.x * blockDim.x + threadIdx.x;
         i < (size_t)KB * KC * KD; i += stride)
        dst[i] = src[i];
}

// out = [centers (B*C*D f32), assign broadcast to (B,N,D) as float values]
__global__ void kmeans_final(const float* __restrict__ centers,
                             const int* __restrict__ assign,
                             float* __restrict__ out)
{
    size_t stride = (size_t)gridDim.x * blockDim.x;
    size_t i0 = (size_t)blockIdx.x * blockDim.x + threadIdx.x;
    const size_t nc = (size_t)KB * KC * KD;
    for (size_t k = i0; k < nc; k += stride) out[k] = centers[k];
    const size_t na = (size_t)KB * KN * KD;
    for (size_t k = i0; k < na; k += stride)
        out[nc + k] = (float)assign[k >> 7];        // k / KD
}

extern "C" void kernel_launch(void* const* d_in, const int* in_sizes, int n_in,
                              void* d_out, int out_size, void* d_ws, size_t ws_size,
                              hipStream_t stream)
{
    const float* data       = (const float*)d_in[0];   // [B,N,D] f32
    const float* centers_in = (const float*)d_in[1];   // [B,C,D] f32
    float* out = (float*)d_out;

    // Workspace layout (~6.05 MB total)
    float* centers = (float*)d_ws;                          // B*C*D
    float* cnorm   = centers + (size_t)KB * KC * KD;        // B*C
    int*   assign  = (int*)(cnorm + (size_t)KB * KC);       // B*N
    float* sums    = (float*)(assign + (size_t)KB * KN);    // B*C*D
    float* counts  = sums + (size_t)KB * KC * KD;           // B*C

    const size_t lds_bytes = (size_t)(KC * KD + KC) * sizeof(float); // ~258 KB

    kmeans_copyc<<<512, 256, 0, stream>>>(centers_in, centers);

    for (int it = 0; it < KITERS; ++it) {
        kmeans_cnorm<<<(KB * KC + 255) / 256, 256, 0, stream>>>(centers, cnorm);

        dim3 ag(KN / 128, KB);
        kmeans_assign<<<ag, 256, lds_bytes, stream>>>(data, centers, cnorm, assign);

        kmeans_zero<<<512, 256, 0, stream>>>(sums, counts);

        dim3 ug(UPD_BLOCKS, KB);
        kmeans_accum<<<ug, 256, lds_bytes, stream>>>(data, assign, sums, counts);

        kmeans_update<<<KB * KC, KD, 0, stream>>>(sums, counts, centers);
    }

    kmeans_final<<<8192, 256, 0, stream>>>(centers, assign, out);
}